// RNAModel_88579405512902
// MI455X (gfx1250) — compile-verified
//
#include <hip/hip_runtime.h>
#include <math.h>

// ---------------------------------------------------------------------------
// Hyperbolic GNN edge/message kernel for gfx1250 (MI455X), fp32 end-to-end.
// GEMMs run on V_WMMA_F32_16X16X4_F32 (one 16-edge tile per wave32).
// Memory-bound problem (~307MB h_E stream vs ~23 GFLOP), so fp32 WMMA is free.
// ---------------------------------------------------------------------------

#define WAVES_PER_BLOCK 2
#define EDGES_PER_WAVE 16
#define ASTRIDE 332            // floats; %4==0 (b128 LDS stores), 332%64=12 -> conflict-free rows
#define EPSF 1e-15f

typedef float v2f __attribute__((ext_vector_type(2)));
typedef float v8f __attribute__((ext_vector_type(8)));

__device__ __forceinline__ float red16(float v) {
  // sum across the 16 lanes sharing (lane>>4)
  v += __shfl_xor(v, 1, 32);
  v += __shfl_xor(v, 2, 32);
  v += __shfl_xor(v, 4, 32);
  v += __shfl_xor(v, 8, 32);
  return v;
}

// One K-step (K=4) against 4 N-tiles (64 output cols). B fragment: lane holds
// col n = nl, rows K = {2*half, 2*half+1} -> contiguous b64 from row-major W[n][k].
__device__ __forceinline__ void wmma_nt4(v8f acc[4], v2f a, const float* __restrict__ W,
                                         int ldk, int kb, int nl) {
#pragma unroll
  for (int nt = 0; nt < 4; ++nt) {
    v2f b = *(const v2f*)(W + (size_t)(nt * 16 + nl) * ldk + kb);
    acc[nt] = __builtin_amdgcn_wmma_f32_16x16x4_f32(false, a, false, b, (short)0, acc[nt],
                                                    false, false);
  }
}

// bias -> expmap0 -> relu -> projx -> logmap0 ; writes next 16x64 A tile to LDS
__device__ __forceinline__ void epilogue_mid(v8f acc[4], const float* __restrict__ Wb,
                                             float* At, int nl, int half, float sc,
                                             float maxn) {
  float bias[4];
#pragma unroll
  for (int nt = 0; nt < 4; ++nt) bias[nt] = Wb[nt * 16 + nl];
  float p[8], pp[8];
#pragma unroll
  for (int i = 0; i < 8; ++i) { p[i] = 0.f; pp[i] = 0.f; }
#pragma unroll
  for (int nt = 0; nt < 4; ++nt) {
#pragma unroll
    for (int i = 0; i < 8; ++i) {
      float v = acc[nt][i] + bias[nt];
      acc[nt][i] = v;
      p[i] += v * v;
      float r = fmaxf(v, 0.f);
      pp[i] += r * r;
    }
  }
  float g[8];
#pragma unroll
  for (int i = 0; i < 8; ++i) {
    float sp = red16(p[i]);
    float spp = red16(pp[i]);
    float n1 = sqrtf(sp);
    float fe = tanhf(sc * n1) / (sc * n1 + EPSF);          // expmap0 factor
    float n2 = fe * sqrtf(spp);                            // |relu(expmap0(u))|
    float pr = fminf(maxn / fmaxf(n2, EPSF), 1.f);         // projx
    float n3 = n2 * pr;
    float fg = atanhf(fminf(sc * n3, 1.f - 1e-5f)) / (sc * (n3 + EPSF)); // logmap0
    g[i] = fe * pr * fg;
  }
#pragma unroll
  for (int nt = 0; nt < 4; ++nt) {
#pragma unroll
    for (int i = 0; i < 8; ++i) {
      int r = i + 8 * half;
      At[r * ASTRIDE + nt * 16 + nl] = g[i] * fmaxf(acc[nt][i], 0.f);
    }
  }
}

__global__ __launch_bounds__(WAVES_PER_BLOCK * 32, 1)
void edge_kernel(const float* __restrict__ hV, const float* __restrict__ hE,
                 const int* __restrict__ src, const int* __restrict__ dst,
                 const float* __restrict__ log_c,
                 const float* __restrict__ W1w, const float* __restrict__ W1b,
                 const float* __restrict__ W2w, const float* __restrict__ W2b,
                 const float* __restrict__ W3w, const float* __restrict__ W3b,
                 float* __restrict__ sums, float* __restrict__ counts, int nEdges) {
  extern __shared__ float smem[];
  const int tid = threadIdx.x;
  const int wave = tid >> 5;
  const int lane = tid & 31;
  const int half = lane >> 4;
  const int nl = lane & 15;

  float* At = smem + (size_t)wave * 16 * ASTRIDE;
  int* sIdx = (int*)(smem + (size_t)WAVES_PER_BLOCK * 16 * ASTRIDE) + wave * 16;

  const float c = __expf(log_c[0]);
  const float sc = sqrtf(c);
  const float maxn = (1.0f - 1e-5f) / sc;

  const int e0 = (blockIdx.x * WAVES_PER_BLOCK + wave) * EDGES_PER_WAVE;

  // ---------------- load phase: 2 lanes per edge, build raw 320-wide A tile --
  const int el = lane >> 1;
  const int part = lane & 1;
  int e = e0 + el;
  if (e >= nEdges) e = nEdges - 1;  // clamp (scatter is guarded below)
  const int s = src[e];
  const int d = dst[e];
  if (part == 0) sIdx[el] = s;

  float ssE = 0.f, ssS = 0.f, ssD = 0.f;
  {
    float* row = At + el * ASTRIDE;
    const float4* pe = (const float4*)(hE + (size_t)e * 192 + part * 96);
#pragma unroll
    for (int j = 0; j < 24; ++j) {
      float4 v = pe[j];
      ssE += v.x * v.x + v.y * v.y + v.z * v.z + v.w * v.w;
      *(float4*)(row + part * 96 + j * 4) = v;
    }
    const float4* ps = (const float4*)(hV + (size_t)s * 64 + part * 32);
#pragma unroll
    for (int j = 0; j < 8; ++j) {
      float4 v = ps[j];
      ssS += v.x * v.x + v.y * v.y + v.z * v.z + v.w * v.w;
      *(float4*)(row + 192 + part * 32 + j * 4) = v;
    }
    const float4* pd = (const float4*)(hV + (size_t)d * 64 + part * 32);
#pragma unroll
    for (int j = 0; j < 8; ++j) {
      float4 v = pd[j];
      ssD += v.x * v.x + v.y * v.y + v.z * v.z + v.w * v.w;
      *(float4*)(row + 256 + part * 32 + j * 4) = v;
    }
  }
  ssE += __shfl_xor(ssE, 1, 32);
  ssS += __shfl_xor(ssS, 1, 32);
  ssD += __shfl_xor(ssD, 1, 32);

  // per-segment projx scales + logmap0 factor over the 320-d concat
  float pE = fminf(maxn / fmaxf(sqrtf(ssE), EPSF), 1.f);
  float pS = fminf(maxn / fmaxf(sqrtf(ssS), EPSF), 1.f);
  float pD = fminf(maxn / fmaxf(sqrtf(ssD), EPSF), 1.f);
  float n320 = sqrtf(pE * pE * ssE + pS * pS * ssS + pD * pD * ssD);
  float fl = atanhf(fminf(sc * n320, 1.f - 1e-5f)) / (sc * (n320 + EPSF));
  float gE = pE * fl, gS = pS * fl, gD = pD * fl;
  // redistribute edge factors: A-fragment row m = nl lives at load lanes {2m,2m+1}
  gE = __shfl(gE, 2 * nl, 32);
  gS = __shfl(gS, 2 * nl, 32);
  gD = __shfl(gD, 2 * nl, 32);

  asm volatile("s_wait_dscnt 0" ::: "memory");

  const float* Am = At + nl * ASTRIDE;  // A-fragment row (lane = M)
  const int kbh = 2 * half;             // K phase per ISA 16x4 f32 A layout

  // ---------------- GEMM1: [16,320] x [320,64], scale folded into A loads ----
  v8f acc[4];
#pragma unroll
  for (int nt = 0; nt < 4; ++nt)
#pragma unroll
    for (int i = 0; i < 8; ++i) acc[nt][i] = 0.f;

#pragma unroll 4
  for (int kk = 0; kk < 48; ++kk) {  // K in [0,192): h_E segment
    int kb = 4 * kk + kbh;
    v2f a = *(const v2f*)(Am + kb);
    a.x *= gE; a.y *= gE;
    wmma_nt4(acc, a, W1w, 320, kb, nl);
  }
#pragma unroll 4
  for (int kk = 48; kk < 64; ++kk) {  // K in [192,256): h_V[src] segment
    int kb = 4 * kk + kbh;
    v2f a = *(const v2f*)(Am + kb);
    a.x *= gS; a.y *= gS;
    wmma_nt4(acc, a, W1w, 320, kb, nl);
  }
#pragma unroll 4
  for (int kk = 64; kk < 80; ++kk) {  // K in [256,320): h_V[dst] segment
    int kb = 4 * kk + kbh;
    v2f a = *(const v2f*)(Am + kb);
    a.x *= gD; a.y *= gD;
    wmma_nt4(acc, a, W1w, 320, kb, nl);
  }
  epilogue_mid(acc, W1b, At, nl, half, sc, maxn);
  asm volatile("s_wait_dscnt 0" ::: "memory");

  // ---------------- GEMM2: [16,64] x [64,64] ----
#pragma unroll
  for (int nt = 0; nt < 4; ++nt)
#pragma unroll
    for (int i = 0; i < 8; ++i) acc[nt][i] = 0.f;
#pragma unroll 4
  for (int kk = 0; kk < 16; ++kk) {
    int kb = 4 * kk + kbh;
    v2f a = *(const v2f*)(Am + kb);
    wmma_nt4(acc, a, W2w, 64, kb, nl);
  }
  epilogue_mid(acc, W2b, At, nl, half, sc, maxn);
  asm volatile("s_wait_dscnt 0" ::: "memory");

  // ---------------- GEMM3: [16,64] x [64,64] ----
#pragma unroll
  for (int nt = 0; nt < 4; ++nt)
#pragma unroll
    for (int i = 0; i < 8; ++i) acc[nt][i] = 0.f;
#pragma unroll 4
  for (int kk = 0; kk < 16; ++kk) {
    int kb = 4 * kk + kbh;
    v2f a = *(const v2f*)(Am + kb);
    wmma_nt4(acc, a, W3w, 64, kb, nl);
  }

  // ---------------- epilogue3: bias -> expmap0 -> logmap0 -> scatter ----
  float bias[4];
#pragma unroll
  for (int nt = 0; nt < 4; ++nt) bias[nt] = W3b[nt * 16 + nl];
  float p[8];
#pragma unroll
  for (int i = 0; i < 8; ++i) p[i] = 0.f;
#pragma unroll
  for (int nt = 0; nt < 4; ++nt)
#pragma unroll
    for (int i = 0; i < 8; ++i) {
      float v = acc[nt][i] + bias[nt];
      acc[nt][i] = v;
      p[i] += v * v;
    }
  float t[8];
  int ss[8];
#pragma unroll
  for (int i = 0; i < 8; ++i) {
    float n = sqrtf(red16(p[i]));
    float fe = tanhf(sc * n) / (sc * n + EPSF);  // expmap0 factor -> h_msg = fe*u
    float nm = fe * n;                           // |h_msg|
    float tl = atanhf(fminf(sc * nm, 1.f - 1e-5f)) / (sc * (nm + EPSF));  // logmap0
    t[i] = fe * tl;
    ss[i] = sIdx[i + 8 * half];
  }
#pragma unroll
  for (int nt = 0; nt < 4; ++nt)
#pragma unroll
    for (int i = 0; i < 8; ++i) {
      int r = i + 8 * half;
      if (e0 + r < nEdges)
        atomicAdd(sums + (size_t)ss[i] * 64 + nt * 16 + nl, t[i] * acc[nt][i]);
    }
  if (nl == 0) {
#pragma unroll
    for (int i = 0; i < 8; ++i) {
      int r = i + 8 * half;
      if (e0 + r < nEdges) atomicAdd(counts + ss[i], 1.0f);
    }
  }
}

// segment-mean -> expmap0 -> mobius_add with projx(h_V) -> projx. 16 lanes/node.
__global__ __launch_bounds__(256, 1)
void node_kernel(const float* __restrict__ hV, const float* __restrict__ sums,
                 const float* __restrict__ counts, const float* __restrict__ log_c,
                 float* __restrict__ out, int nNodes) {
  int tglob = blockIdx.x * blockDim.x + threadIdx.x;
  int n = tglob >> 4;
  int q = tglob & 15;  // 4 features per lane
  if (n >= nNodes) return;
  const float c = __expf(log_c[0]);
  const float sc = sqrtf(c);
  const float maxn = (1.0f - 1e-5f) / sc;

  float cnt = fmaxf(counts[n], 1.0f);
  float inv = 1.0f / cnt;
  float4 s4 = *(const float4*)(sums + (size_t)n * 64 + q * 4);
  float at[4] = {s4.x * inv, s4.y * inv, s4.z * inv, s4.w * inv};
  float sat = at[0] * at[0] + at[1] * at[1] + at[2] * at[2] + at[3] * at[3];
  sat = red16(sat);
  float na = sqrtf(sat);
  float fea = tanhf(sc * na) / (sc * na + EPSF);  // expmap0
  float y[4] = {fea * at[0], fea * at[1], fea * at[2], fea * at[3]};
  float y2 = fea * fea * sat;

  float4 x4 = *(const float4*)(hV + (size_t)n * 64 + q * 4);
  float sx = x4.x * x4.x + x4.y * x4.y + x4.z * x4.z + x4.w * x4.w;
  sx = red16(sx);
  float px = fminf(maxn / fmaxf(sqrtf(sx), EPSF), 1.f);  // projx(h_V)
  float x[4] = {px * x4.x, px * x4.y, px * x4.z, px * x4.w};
  float x2 = px * px * sx;

  float xy = x[0] * y[0] + x[1] * y[1] + x[2] * y[2] + x[3] * y[3];
  xy = red16(xy);

  float A = 1.f + 2.f * c * xy + c * y2;
  float B = 1.f - c * x2;
  float den = 1.f + 2.f * c * xy + c * c * x2 * y2 + EPSF;
  float r[4];
  float sr = 0.f;
#pragma unroll
  for (int k = 0; k < 4; ++k) {
    r[k] = (A * x[k] + B * y[k]) / den;
    sr += r[k] * r[k];
  }
  sr = red16(sr);
  float pr = fminf(maxn / fmaxf(sqrtf(sr), EPSF), 1.f);  // final projx
  float4 o = {r[0] * pr, r[1] * pr, r[2] * pr, r[3] * pr};
  *(float4*)(out + (size_t)n * 64 + q * 4) = o;
}

extern "C" void kernel_launch(void* const* d_in, const int* in_sizes, int n_in,
                              void* d_out, int out_size, void* d_ws, size_t ws_size,
                              hipStream_t stream) {
  const float* hV = (const float*)d_in[0];
  const float* hE = (const float*)d_in[1];
  const int* src = (const int*)d_in[2];
  const int* dst = (const int*)d_in[3];
  const float* log_c = (const float*)d_in[4];
  const float* W1w = (const float*)d_in[5];
  const float* W1b = (const float*)d_in[6];
  const float* W2w = (const float*)d_in[7];
  const float* W2b = (const float*)d_in[8];
  const float* W3w = (const float*)d_in[9];
  const float* W3b = (const float*)d_in[10];
  float* out = (float*)d_out;

  const int nNodes = in_sizes[0] / 64;
  const int nEdges = in_sizes[2];

  float* sums = (float*)d_ws;                       // [nNodes, 64]
  float* counts = sums + (size_t)nNodes * 64;       // [nNodes]
  hipMemsetAsync(d_ws, 0, ((size_t)nNodes * 64 + nNodes) * sizeof(float), stream);

  const size_t smem = (size_t)WAVES_PER_BLOCK * 16 * ASTRIDE * sizeof(float) +
                      (size_t)WAVES_PER_BLOCK * 16 * sizeof(int);
  const int edgesPerBlock = WAVES_PER_BLOCK * EDGES_PER_WAVE;
  const int nBlocks = (nEdges + edgesPerBlock - 1) / edgesPerBlock;
  edge_kernel<<<nBlocks, WAVES_PER_BLOCK * 32, smem, stream>>>(
      hV, hE, src, dst, log_c, W1w, W1b, W2w, W2b, W3w, W3b, sums, counts, nEdges);

  const int nThreads = nNodes * 16;
  node_kernel<<<(nThreads + 255) / 256, 256, 0, stream>>>(hV, sums, counts, log_c, out,
                                                          nNodes);
}